// EzDetectLoss_16939351015940
// MI455X (gfx1250) — compile-verified
//
#include <hip/hip_runtime.h>
#include <hip/hip_bf16.h>
#include <math.h>

typedef float v2f __attribute__((ext_vector_type(2)));
typedef float v8f __attribute__((ext_vector_type(8)));

#define Bq 32
#define Nq 65536
#define Cq 21
#define Mq 16
#define ANCH (Bq * Nq)        /* 2,097,152 anchors */
#define PADC 25               /* LDS row stride (21 classes + pad, odd => bank-friendly) */
#define REC (1 + 6 * Mq)      /* 97 floats per target row */

struct Ctl {
  int pnum;          // count of positive gt records
  int validCnt;      // count of valid gt records (unique anchors assumed)
  unsigned prefix;   // radix-select running prefix
  int kRem;          // remaining rank within current prefix
  unsigned thresh;   // final 32-bit key of K-th smallest score
  int countLess;     // #elements with key < thresh
  int negN;          // #selected negatives with score < 0 (below thresh)
  int tieCnt;        // #elements with key == thresh and score < 0
  int K;             // 3 * pnum
  float tieVal;      // the (unique) tied score value
  int pad_[6];
  unsigned hist[256];
};

// workspace layout (byte offsets)
#define OFF_CTL 0
#define OFF_NEGP 2048                    // 2048 floats: per-block negSum partials
#define OFF_POSCE (OFF_NEGP + 8192)     // 512 floats
#define OFF_BBV (OFF_POSCE + 2048)      // 512 floats
#define OFF_VBITS (OFF_BBV + 2048)      // 65536 u32 valid-anchor bitmap
#define OFF_SCORE (OFF_VBITS + 262144)  // ANCH floats

__device__ __forceinline__ unsigned fkey(float f) {
  unsigned u = __float_as_uint(f);
  return (u & 0x80000000u) ? ~u : (u | 0x80000000u);  // ascending-float order
}

__global__ void init_kernel(unsigned* ctlw, unsigned* vbits) {
  int i = blockIdx.x * blockDim.x + threadIdx.x;
  if (i < 512) ctlw[i] = 0u;
  if (i < (ANCH / 32)) vbits[i] = 0u;
}

__global__ void parse_kernel(const float* __restrict__ target, unsigned* vbits, Ctl* ctl) {
  int tid = threadIdx.x;
  if (tid >= Bq * Mq) return;
  int b = tid >> 4, m = tid & 15;
  const float* rec = target + b * REC;
  int num = (int)rec[0];
  if (m >= num) return;
  int off = 1 + 6 * m;
  int cls = (int)rec[off];
  int k = (int)rec[off + 5];
  k = k < 0 ? 0 : (k > Nq - 1 ? Nq - 1 : k);
  int an = b * Nq + k;
  atomicOr(&vbits[an >> 5], 1u << (an & 31));
  if (cls > 0) atomicAdd(&ctl->pnum, 1);
}

// One wave = 16 anchors. Stage 16x21 tile to LDS, per-lane max, then
// 6x V_WMMA_F32_16X16X4_F32 with B=ones accumulate exact fp32 row-sums of exps.
__global__ void score_kernel(const float* __restrict__ conf,
                             const unsigned* __restrict__ vbits,
                             float* __restrict__ score) {
  __shared__ float lds[8][16 * PADC];
  int lane = threadIdx.x & 31;
  int wave = threadIdx.x >> 5;
  int tile = blockIdx.x * 8 + wave;   // 16 anchors per tile; grid sized exactly
  int g0 = tile * 16;
  const float* src = conf + g0 * Cq;
  float* T = lds[wave];
  for (int i = lane; i < 16 * Cq; i += 32) {
    int a = i / Cq;
    int cc = i - a * Cq;
    T[a * PADC + cc] = src[i];
  }
  if (lane < 16) {
    T[lane * PADC + 21] = -1e30f;
    T[lane * PADC + 22] = -1e30f;
    T[lane * PADC + 23] = -1e30f;
  }
  __syncthreads();

  int a = lane & 15;           // anchor this lane serves in the A-matrix layout
  int uh = (lane >> 4) & 1;    // half-wave => K sub-pair {0,1} vs {2,3}
  const float* row = T + a * PADC;

  float m = row[0];
#pragma unroll
  for (int c = 1; c < 24; ++c) m = fmaxf(m, row[c]);

  v8f acc = {0.f, 0.f, 0.f, 0.f, 0.f, 0.f, 0.f, 0.f};
  v2f bones;
  bones.x = 1.0f;
  bones.y = 1.0f;
#pragma unroll
  for (int t = 0; t < 6; ++t) {
    v2f av;
    av.x = __expf(row[4 * t + 2 * uh] - m);
    av.y = __expf(row[4 * t + 2 * uh + 1] - m);
    // D = A(16x4) * ones(4x16) + C  => every D column holds fp32 row-sums
    acc = __builtin_amdgcn_wmma_f32_16x16x4_f32(false, av, false, bones,
                                                (short)0, acc, false, false);
  }

  // lanes 0..15 hold rowsum(r) in acc[r]; lanes 16..31 hold rowsum(r+8).
  // Writers: lanes 0..7 -> anchors 0..7 (acc[lane]); lanes 24..31 -> anchors 8..15 (acc[lane-24]).
  int li = lane & 7;
  float s0 = (li & 4) ? acc[4] : acc[0];
  float s1 = (li & 4) ? acc[5] : acc[1];
  float s2 = (li & 4) ? acc[6] : acc[2];
  float s3 = (li & 4) ? acc[7] : acc[3];
  float t0 = (li & 2) ? s2 : s0;
  float t1 = (li & 2) ? s3 : s1;
  float ssum = (li & 1) ? t1 : t0;

  bool writer = (lane < 8) || (lane >= 24);
  if (writer) {
    int idx = g0 + a;
    float lse = m + __logf(ssum);
    float c0 = row[0];
    unsigned w = vbits[idx >> 5];
    bool valid = (w >> (idx & 31)) & 1u;
    score[idx] = valid ? 0.0f : (c0 - lse);  // logp[...,0], assigned anchors -> 0
  }
}

__global__ void hist_kernel(const float* __restrict__ score, Ctl* ctl, int p) {
  __shared__ unsigned h[256];
  for (int i = threadIdx.x; i < 256; i += blockDim.x) h[i] = 0u;
  __syncthreads();
  unsigned pref = ctl->prefix;
  int shift = 24 - 8 * p;
  int stride = gridDim.x * blockDim.x;
  for (int i = blockIdx.x * blockDim.x + threadIdx.x; i < ANCH; i += stride) {
    unsigned u = fkey(score[i]);
    unsigned hi = (p == 0) ? 0u : (u >> ((32 - 8 * p) & 31));
    if (p == 0 || hi == pref) atomicAdd(&h[(u >> shift) & 255u], 1u);
  }
  __syncthreads();
  for (int i = threadIdx.x; i < 256; i += blockDim.x)
    if (h[i]) atomicAdd(&ctl->hist[i], h[i]);
}

__global__ void scan_kernel(Ctl* ctl, int p) {
  if (threadIdx.x != 0) return;
  if (p == 0) {
    ctl->K = 3 * ctl->pnum;
    ctl->kRem = ctl->K;
    ctl->prefix = 0u;
  }
  int kRem = ctl->kRem;
  unsigned prefix = ctl->prefix;
  if (kRem > 0) {
    unsigned cum = 0;
    for (int bkt = 0; bkt < 256; ++bkt) {
      unsigned c = ctl->hist[bkt];
      if (cum + c >= (unsigned)kRem) {
        prefix = (prefix << 8) | (unsigned)bkt;
        kRem -= (int)cum;
        break;
      }
      cum += c;
    }
  }
  for (int bkt = 0; bkt < 256; ++bkt) ctl->hist[bkt] = 0u;
  ctl->kRem = kRem;
  ctl->prefix = prefix;
  if (p == 3) ctl->thresh = (ctl->K > 0) ? prefix : 0u;
}

__global__ void select_kernel(const float* __restrict__ score, Ctl* ctl, float* negP) {
  __shared__ float red[256];
  unsigned thresh = ctl->thresh;
  float ns = 0.f;
  int less = 0, nn = 0;
  int stride = gridDim.x * blockDim.x;
  for (int i = blockIdx.x * blockDim.x + threadIdx.x; i < ANCH; i += stride) {
    float f = score[i];
    unsigned u = fkey(f);
    if (u < thresh) {
      ++less;
      if (f < 0.f) { ns += -f; ++nn; }   // CE of negative = -logp0 = -score
    } else if (u == thresh && f < 0.f) {
      int tp = atomicAdd(&ctl->tieCnt, 1);
      if (tp == 0) ctl->tieVal = f;      // all ties share one exact value
    }
  }
  red[threadIdx.x] = ns;
  __syncthreads();
  for (int s = 128; s > 0; s >>= 1) {
    if (threadIdx.x < (unsigned)s) red[threadIdx.x] += red[threadIdx.x + s];
    __syncthreads();
  }
  if (threadIdx.x == 0) negP[blockIdx.x] = red[0];
  if (less) atomicAdd(&ctl->countLess, less);
  if (nn) atomicAdd(&ctl->negN, nn);
}

__global__ void posbbox_kernel(const float* __restrict__ conf,
                               const float* __restrict__ bbox,
                               const float* __restrict__ target,
                               const float* __restrict__ pred, Ctl* ctl,
                               float* posCe, float* bbv) {
  int tid = threadIdx.x;
  if (tid >= Bq * Mq) return;
  posCe[tid] = 0.f;
  bbv[tid] = 0.f;
  int b = tid >> 4, m = tid & 15;
  const float* rec = target + b * REC;
  int num = (int)rec[0];
  if (m >= num) return;
  int off = 1 + 6 * m;
  int cls = (int)rec[off];
  int k = (int)rec[off + 5];
  k = k < 0 ? 0 : (k > Nq - 1 ? Nq - 1 : k);
  int an = b * Nq + k;

  atomicAdd(&ctl->validCnt, 1);
  // bbox regression (every valid record)
  float pb0 = pred[k * 4 + 0], pb1 = pred[k * 4 + 1];
  float pb2 = pred[k * 4 + 2], pb3 = pred[k * 4 + 3];
  float pcx = (pb0 + pb2) * 0.5f, pcy = (pb1 + pb3) * 0.5f;
  float pw = pb2 - pb0, ph = pb3 - pb1;
  float t0 = rec[off + 1], t1 = rec[off + 2], t2 = rec[off + 3], t3 = rec[off + 4];
  float tw = fmaxf(t2 - t0, 1e-8f), th = fmaxf(t3 - t1, 1e-8f);
  float e[4];
  e[0] = ((t0 + t2) * 0.5f - pcx) / pw;
  e[1] = ((t1 + t3) * 0.5f - pcy) / ph;
  e[2] = __logf(tw / pw);
  e[3] = __logf(th / ph);
  const float* bo = bbox + an * 4;
  float s = 0.f;
#pragma unroll
  for (int j = 0; j < 4; ++j) {
    float d = fabsf(bo[j] - e[j]);
    s += (d < 1.0f) ? 0.5f * d * d : (d - 0.5f);
  }
  bbv[tid] = s;

  if (cls > 0) {  // positive: CE against its class
    int cc = cls > Cq - 1 ? Cq - 1 : cls;
    const float* cp = conf + an * Cq;
    float mm = cp[0];
    for (int c = 1; c < Cq; ++c) mm = fmaxf(mm, cp[c]);
    float se = 0.f;
    for (int c = 0; c < Cq; ++c) se += __expf(cp[c] - mm);
    float lse = mm + __logf(se);
    posCe[tid] = lse - cp[cc];
  }
}

__global__ void finalize_kernel(const float* negP, const float* posCe,
                                const float* bbv, Ctl* ctl, float* out) {
  __shared__ float red[256];
  int t = threadIdx.x;
  float a = 0.f;
  for (int i = t; i < 2048; i += 256) a += negP[i];
  red[t] = a;
  __syncthreads();
  for (int s = 128; s > 0; s >>= 1) {
    if (t < s) red[t] += red[t + s];
    __syncthreads();
  }
  float negSum = red[0];
  __syncthreads();
  red[t] = posCe[t] + posCe[t + 256];
  __syncthreads();
  for (int s = 128; s > 0; s >>= 1) {
    if (t < s) red[t] += red[t + s];
    __syncthreads();
  }
  float posS = red[0];
  __syncthreads();
  red[t] = bbv[t] + bbv[t + 256];
  __syncthreads();
  for (int s = 128; s > 0; s >>= 1) {
    if (t < s) red[t] += red[t + s];
    __syncthreads();
  }
  float bbS = red[0];
  if (t == 0) {
    int need = ctl->K - ctl->countLess;
    int tcnt = ctl->tieCnt;
    int take = need < 0 ? 0 : (need < tcnt ? need : tcnt);
    float ns = negSum + (float)take * (-ctl->tieVal);
    int negN = ctl->negN + take;
    int nsel = ctl->pnum + negN;
    if (nsel < 1) nsel = 1;
    out[0] = (posS + ns) / (float)nsel;
    long nbb = (long)ctl->validCnt * 4;
    if (nbb < 1) nbb = 1;
    out[1] = bbS / (float)nbb;
  }
}

extern "C" void kernel_launch(void* const* d_in, const int* in_sizes, int n_in,
                              void* d_out, int out_size, void* d_ws, size_t ws_size,
                              hipStream_t stream) {
  const float* confOut = (const float*)d_in[0];
  const float* bboxOut = (const float*)d_in[1];
  const float* target = (const float*)d_in[2];
  const float* predBoxes = (const float*)d_in[3];
  float* out = (float*)d_out;
  (void)in_sizes; (void)n_in; (void)out_size; (void)ws_size;

  char* ws = (char*)d_ws;
  Ctl* ctl = (Ctl*)(ws + OFF_CTL);
  float* negP = (float*)(ws + OFF_NEGP);
  float* posCe = (float*)(ws + OFF_POSCE);
  float* bbv = (float*)(ws + OFF_BBV);
  unsigned* vbits = (unsigned*)(ws + OFF_VBITS);
  float* score = (float*)(ws + OFF_SCORE);

  init_kernel<<<256, 256, 0, stream>>>((unsigned*)ws, vbits);
  parse_kernel<<<1, 512, 0, stream>>>(target, vbits, ctl);
  score_kernel<<<ANCH / (16 * 8), 256, 0, stream>>>(confOut, vbits, score);
  for (int p = 0; p < 4; ++p) {
    hist_kernel<<<2048, 256, 0, stream>>>(score, ctl, p);
    scan_kernel<<<1, 32, 0, stream>>>(ctl, p);
  }
  select_kernel<<<2048, 256, 0, stream>>>(score, ctl, negP);
  posbbox_kernel<<<1, 512, 0, stream>>>(confOut, bboxOut, target, predBoxes, ctl,
                                        posCe, bbv);
  finalize_kernel<<<1, 256, 0, stream>>>(negP, posCe, bbv, ctl, out);
}